// CAB_43739946943483
// MI455X (gfx1250) — compile-verified
//
#include <hip/hip_runtime.h>
#include <hip/hip_bf16.h>

typedef float v2f __attribute__((ext_vector_type(2)));
typedef float v4f __attribute__((ext_vector_type(4)));
typedef float v8f __attribute__((ext_vector_type(8)));

// ---------------------------------------------------------------------------
// Kernel A: global average pools (RT loads: warm L2 for the fuse pass).
// Blocks 0..4095 -> shallower rows (4096 floats each) -> gp[b*512 + c]
// Blocks 4096..8191 -> deeper rows (1024 floats each) -> gp[b*512 + 256 + c]
// ---------------------------------------------------------------------------
__global__ void __launch_bounds__(256) pool_kernel(const float* __restrict__ shallower,
                                                   const float* __restrict__ deeper,
                                                   float* __restrict__ gp) {
    const int bid = blockIdx.x;
    const bool isShallow = bid < 4096;
    const int row = isShallow ? bid : bid - 4096;       // row = b*256 + c
    const int n = isShallow ? 4096 : 1024;
    const float* src = isShallow ? (shallower + (size_t)row * 4096)
                                 : (deeper + (size_t)row * 1024);

    float sum = 0.0f;
    for (int i = threadIdx.x * 4; i < n; i += 256 * 4) {
        const v4f v = *(const v4f*)(src + i);
        sum += v.x + v.y + v.z + v.w;
    }
    __shared__ float red[256];
    red[threadIdx.x] = sum;
    __syncthreads();
    for (int s = 128; s > 0; s >>= 1) {
        if (threadIdx.x < s) red[threadIdx.x] += red[threadIdx.x + s];
        __syncthreads();
    }
    if (threadIdx.x == 0) {
        const int b = row >> 8;
        const int c = row & 255;
        gp[b * 512 + (isShallow ? c : 256 + c)] = red[0] / (float)n;
    }
}

// ---------------------------------------------------------------------------
// Kernel B: tiny MLP on the matrix pipe (fp32 WMMA 16x16x4).
// 1 block, 16 waves; wave w owns output channels [16w, 16w+16).
//   h    = relu(gp[16x512] @ w1^T + b1)      (128 WMMAs per wave)
//   gate = h @ w2^T + b2 ; sig = sigmoid     (64 WMMAs per wave)
// A-matrix layout: lane l, vgpr v  -> A[l%16][k0 + v + 2*(l/16)]
// B-matrix layout: lane l, vgpr v  -> B[k0 + v + 2*(l/16)][l%16]
// C/D layout:      lane l, vgpr r  -> D[r + 8*(l/16)][l%16]
// ---------------------------------------------------------------------------
__global__ void __launch_bounds__(512) mlp_kernel(const float* __restrict__ gp,
                                                  const float* __restrict__ w1,
                                                  const float* __restrict__ b1,
                                                  const float* __restrict__ w2,
                                                  const float* __restrict__ b2,
                                                  float* __restrict__ sig) {
    __shared__ float hbuf[16 * 256];
    const int lane = threadIdx.x & 31;
    const int wv   = threadIdx.x >> 5;   // 0..15
    const int half = lane >> 4;          // 0 or 1
    const int nl   = lane & 15;          // A row M / B,D column N
    const int n0   = wv * 16;

    // ---- GEMM1: [16x512] x [512 x 16-slice] ----
    v8f acc = {};
    {
        const float* arow  = gp + (size_t)nl * 512;
        const float* w1row = w1 + (size_t)(n0 + nl) * 512;   // w1[o][i], o = n0+nl
        for (int k0 = 0; k0 < 512; k0 += 4) {
            const int k = k0 + 2 * half;
            v2f a, b;
            a.x = arow[k];  a.y = arow[k + 1];
            b.x = w1row[k]; b.y = w1row[k + 1];
            acc = __builtin_amdgcn_wmma_f32_16x16x4_f32(
                false, a, false, b, (short)0, acc, false, false);
        }
    }
    {
        const float bias1 = b1[n0 + nl];
        #pragma unroll
        for (int r = 0; r < 8; ++r) {
            const int mrow = r + 8 * half;
            const float v = acc[r] + bias1;
            hbuf[mrow * 256 + n0 + nl] = v > 0.0f ? v : 0.0f;   // ReLU
        }
    }
    __syncthreads();

    // ---- GEMM2: [16x256] x [256 x 16-slice] ----
    v8f acc2 = {};
    {
        const float* arow  = hbuf + (size_t)nl * 256;
        const float* w2row = w2 + (size_t)(n0 + nl) * 256;
        for (int k0 = 0; k0 < 256; k0 += 4) {
            const int k = k0 + 2 * half;
            v2f a, b;
            a.x = arow[k];  a.y = arow[k + 1];
            b.x = w2row[k]; b.y = w2row[k + 1];
            acc2 = __builtin_amdgcn_wmma_f32_16x16x4_f32(
                false, a, false, b, (short)0, acc2, false, false);
        }
    }
    {
        const float bias2 = b2[n0 + nl];
        #pragma unroll
        for (int r = 0; r < 8; ++r) {
            const int mrow = r + 8 * half;
            const float gate = acc2[r] + bias2;
            sig[mrow * 256 + n0 + nl] = 1.0f / (1.0f + __expf(-gate));
        }
    }
}

// ---------------------------------------------------------------------------
// Kernel C: out = shallower * sigmoid(gate)[b,c] + bilinear_up2x(deeper).
// One thread -> 4 consecutive x outputs (float4 / B128 on shallower + out).
// shallower loads are last-use -> NT hint; out stores are streaming -> NT,
// so the 192MB L2 keeps deeper/shallower resident instead of the output.
// Half-pixel bilinear 32->64:  x=2i   -> 0.25*d[i-1] + 0.75*d[i]
//                              x=2i+1 -> 0.75*d[i]   + 0.25*d[i+1]   (edges clamp)
// ---------------------------------------------------------------------------
__global__ void __launch_bounds__(256) fuse_kernel(const float* __restrict__ shallower,
                                                   const float* __restrict__ deeper,
                                                   const float* __restrict__ sig,
                                                   float* __restrict__ out) {
    const int idx = blockIdx.x * 256 + threadIdx.x;   // float4 group id
    const int xg = idx & 15;          // x-group: covers x = 4*xg .. 4*xg+3
    const int t  = idx >> 4;
    const int y  = t & 63;
    const int bc = t >> 6;            // b*256 + c

    // row interpolation
    const float srcy = y * 0.5f - 0.25f;
    int y0 = (int)floorf(srcy);
    const float wy = srcy - (float)y0;
    int y1 = y0 + 1;
    y0 = y0 < 0 ? 0 : y0;
    y1 = y1 > 31 ? 31 : y1;
    const float* drow0 = deeper + ((size_t)bc * 32 + y0) * 32;
    const float* drow1 = deeper + ((size_t)bc * 32 + y1) * 32;

    const int j2  = xg * 2;                       // deeper col base
    const int xm1 = (j2 - 1) < 0 ? 0 : (j2 - 1);
    const int xp2 = (j2 + 2) > 31 ? 31 : (j2 + 2);

    const float rm1 = drow0[xm1]    + wy * (drow1[xm1]    - drow0[xm1]);
    const float r0  = drow0[j2]     + wy * (drow1[j2]     - drow0[j2]);
    const float r1  = drow0[j2 + 1] + wy * (drow1[j2 + 1] - drow0[j2 + 1]);
    const float r2  = drow0[xp2]    + wy * (drow1[xp2]    - drow0[xp2]);

    const float g = sig[bc];
    const v4f s = __builtin_nontemporal_load((const v4f*)(shallower + (size_t)idx * 4));
    v4f o;
    o.x = s.x * g + (0.25f * rm1 + 0.75f * r0);
    o.y = s.y * g + (0.75f * r0  + 0.25f * r1);
    o.z = s.z * g + (0.25f * r0  + 0.75f * r1);
    o.w = s.w * g + (0.75f * r1  + 0.25f * r2);
    __builtin_nontemporal_store(o, (v4f*)(out + (size_t)idx * 4));
}

extern "C" void kernel_launch(void* const* d_in, const int* in_sizes, int n_in,
                              void* d_out, int out_size, void* d_ws, size_t ws_size,
                              hipStream_t stream) {
    const float* deeper    = (const float*)d_in[0];  // [16,256,32,32]
    const float* shallower = (const float*)d_in[1];  // [16,256,64,64]
    const float* w1        = (const float*)d_in[2];  // [256,512]
    const float* b1        = (const float*)d_in[3];  // [256]
    const float* w2        = (const float*)d_in[4];  // [256,256]
    const float* b2        = (const float*)d_in[5];  // [256]
    float* out = (float*)d_out;

    float* gp  = (float*)d_ws;   // [16,512] pooled+concat
    float* sig = gp + 16 * 512;  // [16,256] sigmoid gates

    pool_kernel<<<8192, 256, 0, stream>>>(shallower, deeper, gp);
    mlp_kernel<<<1, 512, 0, stream>>>(gp, w1, b1, w2, b2, sig);
    fuse_kernel<<<16384, 256, 0, stream>>>(shallower, deeper, sig, out);
}